// GCNConvNet_52853867545090
// MI455X (gfx1250) — compile-verified
//
#include <hip/hip_runtime.h>
#include <hip/hip_bf16.h>
#include <math.h>

// ---------------------------------------------------------------------------
// GCN (8 layers, HID=64) for MI455X / gfx1250.
//   - FP32 WMMA (v_wmma_f32_16x16x4_f32) for the 64x64 hidden GEMMs,
//     weights staged in LDS.
//   - Aggregation via per-launch CSR build (count + scan + bucket, ~6.6M
//     atomics once) then atomic-free gather-reduce per layer: the 25.6 MB
//     feature table is L2-resident (192 MB L2), so gathers run at L2 rate
//     and outputs are plain coalesced stores (vs ~1.5G f32 atomics for the
//     naive scatter formulation -> atomic-throughput bound).
// ---------------------------------------------------------------------------

typedef float v2f __attribute__((ext_vector_type(2)));
typedef float v8f __attribute__((ext_vector_type(8)));

#define HIDF 64

// ----------------------------- CSR construction ----------------------------

__global__ void k_zero(int* __restrict__ p, int n) {
  int i = blockIdx.x * blockDim.x + threadIdx.x;
  if (i < n) p[i] = 0;
}

__global__ void k_count(const int* __restrict__ dst, int* __restrict__ cnt, int nE) {
  int e = blockIdx.x * blockDim.x + threadIdx.x;
  if (e < nE) atomicAdd(&cnt[dst[e]], 1);
}

// Per-block (256-wide) inclusive Hillis-Steele scan; block totals to bsum.
__global__ void k_scan1(const int* __restrict__ cnt, int* __restrict__ incl,
                        int* __restrict__ bsum, int n) {
  __shared__ int s[256];
  int i = blockIdx.x * 256 + threadIdx.x;
  s[threadIdx.x] = (i < n) ? cnt[i] : 0;
  __syncthreads();
#pragma unroll
  for (int off = 1; off < 256; off <<= 1) {
    int t = (threadIdx.x >= off) ? s[threadIdx.x - off] : 0;
    __syncthreads();
    s[threadIdx.x] += t;
    __syncthreads();
  }
  if (i < n) incl[i] = s[threadIdx.x];
  if (threadIdx.x == 255) bsum[blockIdx.x] = s[255];
}

// Serial exclusive scan of ~391 block sums (single lane, negligible).
__global__ void k_scan2(int* __restrict__ bsum, int nb) {
  int acc = 0;
  for (int i = 0; i < nb; ++i) { int v = bsum[i]; bsum[i] = acc; acc += v; }
}

// rowStart[i] = exclusive scan; also writes rowStart[n] = total.
__global__ void k_scan3(const int* __restrict__ cnt, const int* __restrict__ incl,
                        const int* __restrict__ bsum, int* __restrict__ rowStart, int n) {
  int i = blockIdx.x * 256 + threadIdx.x;
  if (i >= n) return;
  int base = bsum[i >> 8];
  rowStart[i] = incl[i] - cnt[i] + base;
  if (i == n - 1) rowStart[n] = incl[i] + base;
}

__global__ void k_dinv(const int* __restrict__ cnt, float* __restrict__ dinv, int n) {
  int i = blockIdx.x * blockDim.x + threadIdx.x;
  if (i < n) dinv[i] = rsqrtf((float)(cnt[i] + 1));  // +1: self-loop
}

__global__ void k_copy(const int* __restrict__ a, int* __restrict__ b, int n) {
  int i = blockIdx.x * blockDim.x + threadIdx.x;
  if (i < n) b[i] = a[i];
}

// Bucket edges by dst: eSW[pos] = {src, dinv[src]*dinv[dst]} packed in 8B.
__global__ void k_bucket(const int* __restrict__ src, const int* __restrict__ dst,
                         const float* __restrict__ dinv, int* __restrict__ cursor,
                         int2* __restrict__ eSW, int nE) {
  int e = blockIdx.x * blockDim.x + threadIdx.x;
  if (e >= nE) return;
  int s = src[e], d = dst[e];
  int pos = atomicAdd(&cursor[d], 1);
  eSW[pos] = make_int2(s, __float_as_int(dinv[s] * dinv[d]));
}

// ------------------------------- GEMM kernels ------------------------------

// Input layer: T[n,f] = sum_{k<3} x[n,k] * Win[k,f]   (K=3, scalar FMA)
__global__ void k_gemm_in(const float* __restrict__ x, const float* __restrict__ Win,
                          float* __restrict__ T, int nNodes) {
  int t = blockIdx.x * blockDim.x + threadIdx.x;
  if (t >= nNodes * HIDF) return;
  int n = t >> 6, f = t & 63;
  const float* xr = x + n * 3;
  T[t] = fmaf(xr[0], Win[f], fmaf(xr[1], Win[64 + f], xr[2] * Win[128 + f]));
}

// Hidden GEMM: T = relu(X) @ W, X:[N,64], W:[64,64] row-major, FP32 WMMA.
// One wave computes one 16x16 output tile; 4 waves/block cover all 4 N-tiles.
// A frag (16x4 f32): lane l -> row m=l&15, K-pair base kh=(l>>4)*2, 2 floats.
// B frag (4x16 f32): lane l -> col n=l&15, same K-pair split across half-waves.
// C/D (16x16 f32, 8 VGPRs): elem[v] -> m = v + 8*(l>>4), n = l&15.
__global__ void __launch_bounds__(128)
k_gemm_wmma(const float* __restrict__ Xin, const float* __restrict__ W,
            float* __restrict__ T) {
  __shared__ float sW[HIDF * HIDF];  // 16 KB
  for (int i = threadIdx.x; i < HIDF * HIDF; i += 128) sW[i] = W[i];
  __syncthreads();

  const int wave  = threadIdx.x >> 5;
  const int lane  = threadIdx.x & 31;
  const int mbase = blockIdx.x * 16;
  const int nbase = wave * 16;
  const int mrow  = lane & 15;
  const int kh    = (lane >> 4) * 2;

  const float* __restrict__ arow = Xin + (size_t)(mbase + mrow) * HIDF;
  v8f c = {};

#pragma unroll
  for (int k = 0; k < HIDF; k += 4) {
    v2f a, b;
    a.x = fmaxf(arow[k + kh], 0.0f);          // fused ReLU on A load
    a.y = fmaxf(arow[k + kh + 1], 0.0f);
    b.x = sW[(k + kh) * HIDF + nbase + mrow];
    b.y = sW[(k + kh + 1) * HIDF + nbase + mrow];
    // 8 args: (neg_a, A, neg_b, B, c_mod, C, reuse_a, reuse_b)
    c = __builtin_amdgcn_wmma_f32_16x16x4_f32(false, a, false, b, (short)0, c,
                                              false, false);
  }

  const int n  = nbase + mrow;
  const int mo = (lane >> 4) * 8;
#pragma unroll
  for (int v = 0; v < 8; ++v)
    T[(size_t)(mbase + mo + v) * HIDF + n] = c[v];
}

// Output layer: T6[n,j] = sum_k relu(X[n,k]) * Wout[k,j], Wout:[64,6]
__global__ void k_gemm_out(const float* __restrict__ X, const float* __restrict__ Wout,
                           float* __restrict__ T6, int nNodes) {
  int t = blockIdx.x * blockDim.x + threadIdx.x;
  if (t >= nNodes * 6) return;
  int n = t / 6, j = t % 6;
  const float* xr = X + (size_t)n * HIDF;
  float acc = 0.0f;
#pragma unroll 8
  for (int k = 0; k < HIDF; ++k) acc = fmaf(fmaxf(xr[k], 0.0f), Wout[k * 6 + j], acc);
  T6[t] = acc;
}

// ----------------------------- gather-aggregate ----------------------------

// X[n,f] = dinv[n]^2*T[n,f] + b[f] + sum_{in-edges} w * T[src,f]
// Block 256 = 4 nodes x 64 features; src/w uniform per wave (L0 broadcast).
__global__ void k_gather64(const int* __restrict__ rowStart, const int2* __restrict__ eSW,
                           const float* __restrict__ dinv, const float* __restrict__ T,
                           const float* __restrict__ b, float* __restrict__ X, int nNodes) {
  int t = blockIdx.x * blockDim.x + threadIdx.x;
  if (t >= nNodes * HIDF) return;
  int n = t >> 6, f = t & 63;
  float di  = dinv[n];
  float acc = fmaf(di * di, T[(size_t)n * HIDF + f], b[f]);  // self-loop + bias
  int beg = rowStart[n], end = rowStart[n + 1];
  for (int i = beg; i < end; ++i) {
    int2 e = eSW[i];
    acc = fmaf(__int_as_float(e.y), T[(size_t)e.x * HIDF + f], acc);
  }
  X[t] = acc;
}

// Final 6-wide gather with fused bias + sigmoid. 8 threads/node (6 active).
__global__ void k_gather6(const int* __restrict__ rowStart, const int2* __restrict__ eSW,
                          const float* __restrict__ dinv, const float* __restrict__ T6,
                          const float* __restrict__ b, float* __restrict__ O, int nNodes) {
  int t = blockIdx.x * blockDim.x + threadIdx.x;
  int n = t >> 3, f = t & 7;
  if (n >= nNodes || f >= 6) return;
  float di  = dinv[n];
  float acc = fmaf(di * di, T6[(size_t)n * 6 + f], b[f]);
  int beg = rowStart[n], end = rowStart[n + 1];
  for (int i = beg; i < end; ++i) {
    int2 e = eSW[i];
    acc = fmaf(__int_as_float(e.y), T6[(size_t)e.x * 6 + f], acc);
  }
  O[(size_t)n * 6 + f] = 1.0f / (1.0f + expf(-acc));
}

static inline int cdiv(long long a, int b) { return (int)((a + b - 1) / b); }

extern "C" void kernel_launch(void* const* d_in, const int* in_sizes, int n_in,
                              void* d_out, int out_size, void* d_ws, size_t ws_size,
                              hipStream_t stream) {
  const float* x     = (const float*)d_in[0];   // [N,3]
  const int*   eidx  = (const int*)d_in[1];     // [2,E]
  const float* W_in  = (const float*)d_in[2];   // [3,64]
  const float* b_in  = (const float*)d_in[3];   // [64]
  const float* W_hid = (const float*)d_in[4];   // [6,64,64]
  const float* b_hid = (const float*)d_in[5];   // [6,64]
  const float* W_out = (const float*)d_in[6];   // [64,6]
  const float* b_out = (const float*)d_in[7];   // [6]
  float* Out = (float*)d_out;                   // [N,6]

  const int N = in_sizes[0] / 3;
  const int E = in_sizes[1] / 2;
  const int* src = eidx;
  const int* dst = eidx + E;
  const int nb = cdiv(N, 256);

  // Workspace layout (~81 MB):
  char* w = (char*)d_ws;
  auto take = [&](size_t bytes) { char* p = w; w += (bytes + 255) & ~(size_t)255; return p; };
  int*   cnt      = (int*)  take((size_t)N * 4);
  int*   incl     = (int*)  take((size_t)N * 4);
  int*   bsum     = (int*)  take((size_t)nb * 4);
  int*   rowStart = (int*)  take((size_t)(N + 1) * 4);
  int*   cursor   = (int*)  take((size_t)N * 4);
  float* dinv     = (float*)take((size_t)N * 4);
  int2*  eSW      = (int2*) take((size_t)E * 8);
  float* X        = (float*)take((size_t)N * HIDF * 4);
  float* T        = (float*)take((size_t)N * HIDF * 4);
  float* T6       = (float*)take((size_t)N * 6 * 4);

  const int B = 256;

  // --- CSR build + symmetric normalization (stateless: rebuilt every call) ---
  k_zero  <<<cdiv(N, B), B, 0, stream>>>(cnt, N);
  k_count <<<cdiv(E, B), B, 0, stream>>>(dst, cnt, E);
  k_scan1 <<<nb, 256, 0, stream>>>(cnt, incl, bsum, N);
  k_scan2 <<<1, 1, 0, stream>>>(bsum, nb);
  k_scan3 <<<nb, 256, 0, stream>>>(cnt, incl, bsum, rowStart, N);
  k_dinv  <<<cdiv(N, B), B, 0, stream>>>(cnt, dinv, N);
  k_copy  <<<cdiv(N, B), B, 0, stream>>>(rowStart, cursor, N);
  k_bucket<<<cdiv(E, B), B, 0, stream>>>(src, dst, dinv, cursor, eSW, E);

  // --- input layer (K=3) ---
  k_gemm_in  <<<cdiv((long long)N * HIDF, B), B, 0, stream>>>(x, W_in, T, N);
  k_gather64 <<<cdiv((long long)N * HIDF, B), B, 0, stream>>>(rowStart, eSW, dinv, T, b_in, X, N);

  // --- 6 hidden layers: WMMA GEMM (fused ReLU on load) + gather ---
  for (int i = 0; i < 6; ++i) {
    k_gemm_wmma<<<N / 16, 128, 0, stream>>>(X, W_hid + (size_t)i * HIDF * HIDF, T);
    k_gather64 <<<cdiv((long long)N * HIDF, B), B, 0, stream>>>(
        rowStart, eSW, dinv, T, b_hid + (size_t)i * HIDF, X, N);
  }

  // --- output layer (64 -> 6) with fused bias + sigmoid in the gather ---
  k_gemm_out<<<cdiv((long long)N * 6, B), B, 0, stream>>>(X, W_out, T6, N);
  k_gather6 <<<cdiv((long long)N * 8, B), B, 0, stream>>>(rowStart, eSW, dinv, T6, b_out, Out, N);
}